// Decoding_model_30820685316188
// MI455X (gfx1250) — compile-verified
//
#include <hip/hip_runtime.h>

#define M_CHECKS 63
#define N_VARS   127
#define BATCH    256
#define N_ITERS  3
#define BIGF     1e30f
#define HTOT     (M_CHECKS * N_VARS)   // 8001

// Async global->LDS staging (CDNA5 ASYNCcnt path), guarded so a missing builtin
// falls back to plain loads instead of breaking the build.
#if defined(__gfx1250__) && __has_builtin(__builtin_amdgcn_global_load_async_to_lds_b32) && __has_builtin(__builtin_amdgcn_s_wait_asynccnt)
#define USE_ASYNC_LDS 1
#endif

__device__ __forceinline__ float sgnf(float x) {
    return (float)((x > 0.0f) - (x < 0.0f));
}

__device__ __forceinline__ float wave_sum(float v) {
    #pragma unroll
    for (int off = 16; off > 0; off >>= 1) v += __shfl_xor(v, off, 32);
    return v;
}

__launch_bounds__(256)
__global__ void ldpc_decode_kernel(const float* __restrict__ soft_in,
                                   const int*   __restrict__ labels,
                                   const float* __restrict__ Hmat,
                                   const float* __restrict__ check_w,
                                   float* __restrict__ outs,        // [4, B, N]
                                   float* __restrict__ loss_part)   // [B]
{
    __shared__ float sH[HTOT];             // H staged flat (only used to build reg masks)
    __shared__ float sx[128];              // current soft row
    __shared__ float sacc[8][128];         // per-wave bijective scatter lines
    __shared__ float swsum[8];

    const int b    = blockIdx.x;
    const int tid  = threadIdx.x;
    const int wave = tid >> 5;
    const int lane = tid & 31;

    __builtin_prefetch(&Hmat[0], 0, 1);                 // global_prefetch_b8
    __builtin_prefetch(&soft_in[b * N_VARS], 0, 1);

    // ---- stage H into LDS via async copy engine ----
#ifdef USE_ASYNC_LDS
    for (int i = tid; i < HTOT; i += 256) {
        float* gp = const_cast<float*>(&Hmat[i]);
        __builtin_amdgcn_global_load_async_to_lds_b32(
            (__attribute__((address_space(1))) int*)gp,
            (__attribute__((address_space(3))) int*)&sH[i],
            0, 0);
    }
    __builtin_amdgcn_s_wait_asynccnt(0);
#else
    for (int i = tid; i < HTOT; i += 256) sH[i] = Hmat[i];
#endif

    // ---- load soft row, emit outs[0] (copy of input) ----
    if (tid < N_VARS) {
        float v = soft_in[b * N_VARS + tid];
        sx[tid] = v;
        outs[b * N_VARS + tid] = v;
    }

    // softplus(check_weight)
    const float cw = check_w[0];
    const float W  = fmaxf(cw, 0.0f) + log1pf(expf(-fabsf(cw)));

    // ---- combo setup: wave -> (transform, shift) ----
    // split == perm (both are n -> x[2n mod 127]); perm branch weighted x2.
    const int shifts4[4] = {0, 31, 62, 93};
    const int trans = wave >> 2;                 // 0: identity, 1: x2 permutation
    const int shift = shifts4[wave & 3];
    const float Wscale = W * (trans ? 2.0f : 1.0f);

    int  nIdx[4], srcIdx[4];
    bool valid[4];
    #pragma unroll
    for (int k = 0; k < 4; ++k) {
        int n = lane + 32 * k;
        nIdx[k]  = n;
        valid[k] = (n < N_VARS);
        int j = n - shift; if (j < 0) j += N_VARS;
        if (!valid[k]) j = 0;
        srcIdx[k] = trans ? ((2 * j) % N_VARS) : j;   // same map for gather & scatter
    }

    __syncthreads();

    // ---- build per-lane H bitmasks in registers: bit m of hbits[k] = H[m][n_k] ----
    // After this, the hot loop touches no LDS for H at all.
    unsigned long long hbits[4] = {0ull, 0ull, 0ull, 0ull};
    for (int m = 0; m < M_CHECKS; ++m) {
        #pragma unroll
        for (int k = 0; k < 4; ++k)
            if (valid[k] && (sH[m * N_VARS + nIdx[k]] != 0.0f))
                hbits[k] |= (1ull << m);
    }

    // ---- 3 BP iterations, register/LDS resident ----
    for (int it = 0; it < N_ITERS; ++it) {
        float y[4], av[4], sv[4];
        #pragma unroll
        for (int k = 0; k < 4; ++k) {
            y[k]  = valid[k] ? sx[srcIdx[k]] : 0.0f;
            av[k] = fminf(fabsf(y[k]), BIGF);    // clip(|vc|, 0, BIG)
            sv[k] = sgnf(y[k]);
        }

        float acc[4] = {0.0f, 0.0f, 0.0f, 0.0f};
        unsigned long long mm[4] = {hbits[0], hbits[1], hbits[2], hbits[3]};

        for (int m = 0; m < M_CHECKS; ++m) {
            bool on[4];
            float m1 = BIGF, m2 = BIGF;
            bool negpar = false;   // parity of masked negatives (this lane)
            bool haszero = false;  // any masked exact zero (this lane)
            #pragma unroll
            for (int k = 0; k < 4; ++k) {
                bool h = (mm[k] & 1ull) != 0ull;
                mm[k] >>= 1;
                on[k] = h;
                if (h) {
                    negpar  ^= (y[k] < 0.0f);
                    haszero |= (y[k] == 0.0f);
                    float a = av[k];
                    if (a < m1) { m2 = m1; m1 = a; }
                    else if (a < m2) { m2 = a; }
                }
            }
            // wave32 merge of (min1,min2)
            #pragma unroll
            for (int off = 16; off > 0; off >>= 1) {
                float o1 = __shfl_xor(m1, off, 32);
                float o2 = __shfl_xor(m2, off, 32);
                float lo = fminf(m1, o1);
                float hi = fmaxf(m1, o1);
                m2 = fminf(hi, fminf(m2, o2));
                m1 = lo;
            }
            // row sign product via ballots (exact: in {-1, 0, +1})
            unsigned long long nb = __ballot(negpar);
            unsigned long long zb = __ballot(haszero);
            float prod = (zb != 0ull) ? 0.0f
                       : ((__popcll(nb) & 1) ? -1.0f : 1.0f);
            // check->variable messages for this check row
            #pragma unroll
            for (int k = 0; k < 4; ++k) {
                if (on[k]) {
                    float val = (av[k] > m1) ? m1 : m2;   // min-sum extrinsic select
                    acc[k] += val * sv[k] * prod;
                }
            }
        }

        // bijective scatter: each wave writes every slot [0,127) exactly once
        #pragma unroll
        for (int k = 0; k < 4; ++k)
            if (valid[k]) sacc[wave][srcIdx[k]] = acc[k] * Wscale;
        __syncthreads();

        // fixed-order combine (deterministic), update soft, emit outs[it+1]
        if (tid < N_VARS) {
            float s = 0.0f;
            #pragma unroll
            for (int w = 0; w < 8; ++w) s += sacc[w][tid];
            float nv = sx[tid] + s * (1.0f / 12.0f);   // /(3*NUM_SHIFTS)
            sx[tid] = nv;
            outs[(size_t)(it + 1) * BATCH * N_VARS + (size_t)b * N_VARS + tid] = nv;
        }
        __syncthreads();
    }

    // ---- per-row loss partial (weighted BCE on final soft) ----
    float lv = 0.0f;
    if (tid < N_VARS) {
        float x  = sx[tid];
        float lf = (float)labels[b * N_VARS + tid];
        float target = 1.0f - 2.0f * lf;
        float w  = (sgnf(x) != target) ? 2.0f : 1.0f;
        float z  = -x;
        float ce = fmaxf(z, 0.0f) - z * lf + log1pf(expf(-fabsf(z)));
        lv = w * ce;
    }
    lv = wave_sum(lv);
    if (lane == 0) swsum[wave] = lv;
    __syncthreads();
    if (wave == 0) {
        float v = (lane < 8) ? swsum[lane] : 0.0f;
        v = wave_sum(v);
        if (lane == 0) loss_part[b] = v;
    }
}

__launch_bounds__(256)
__global__ void loss_reduce_kernel(const float* __restrict__ part,
                                   float* __restrict__ out_loss)
{
    __shared__ float sw[8];
    const int tid = threadIdx.x, wave = tid >> 5, lane = tid & 31;
    float v = part[tid];
    v = wave_sum(v);
    if (lane == 0) sw[wave] = v;
    __syncthreads();
    if (wave == 0) {
        float x = (lane < 8) ? sw[lane] : 0.0f;
        x = wave_sum(x);
        if (lane == 0) *out_loss = x;
    }
}

extern "C" void kernel_launch(void* const* d_in, const int* in_sizes, int n_in,
                              void* d_out, int out_size, void* d_ws, size_t ws_size,
                              hipStream_t stream) {
    const float* soft   = (const float*)d_in[0];
    const int*   labels = (const int*)  d_in[1];
    const float* H      = (const float*)d_in[2];
    const float* cw     = (const float*)d_in[3];
    float* out = (float*)d_out;
    float* ws  = (float*)d_ws;    // 256 per-row loss partials

    ldpc_decode_kernel<<<BATCH, 256, 0, stream>>>(soft, labels, H, cw, out, ws);
    loss_reduce_kernel<<<1, 256, 0, stream>>>(ws, out + (size_t)4 * BATCH * N_VARS);
}